// MSEObserver_59322088292792
// MI455X (gfx1250) — compile-verified
//
#include <hip/hip_runtime.h>
#include <hip/hip_bf16.h>

// ---------------- configuration ----------------
#define NELEM   4194304   // 1024*4096
#define NBLK    512       // scoring blocks
#define TILE    8192      // elements per block tile (32 KB LDS)
#define NCAND   1600      // 100 steps * 16 zero-points
#define CPT     7         // candidates per thread (256*7 >= 1600)
#define EPSF    1.1920928955078125e-07f

// workspace layout (float offsets)
#define OFF_BMIN   0        // 512 block mins
#define OFF_BMAX   512      // 512 block maxs
#define OFF_MM     1024     // [xmin, xmax]
#define OFF_S      1056     // 1600 scales
#define OFF_Z      2656     // 1600 zero points
#define OFF_SCORES 4256     // 1600 scores
#define OFF_PART   6144     // 1600 x 512 partial sums

typedef __attribute__((ext_vector_type(2))) float v2f;
typedef __attribute__((ext_vector_type(8))) float v8f;
typedef __attribute__((ext_vector_type(4))) int   v4i;
typedef __attribute__((address_space(1))) v4i     gv4i;   // global-space 16B vector
typedef __attribute__((address_space(3))) v4i     lv4i;   // LDS-space 16B vector

#if defined(__has_builtin)
#if __has_builtin(__builtin_amdgcn_global_load_async_to_lds_b128)
#define HAVE_ASYNC_LDS 1
#endif
#if __has_builtin(__builtin_amdgcn_wmma_f32_16x16x4_f32)
#define HAVE_WMMA_F32X4 1
#endif
#endif

// ---------------- kernel 1: per-block min/max ----------------
__global__ __launch_bounds__(256) void minmax_part(const float* __restrict__ x,
                                                   float* __restrict__ wsmin,
                                                   float* __restrict__ wsmax) {
    __shared__ float smn[256], smx[256];
    int t = threadIdx.x;
    const float4* x4 = (const float4*)(x + (size_t)blockIdx.x * TILE);
    float mn = INFINITY, mx = -INFINITY;
#pragma unroll
    for (int v = 0; v < 8; ++v) {
        float4 f = x4[v * 256 + t];
        mn = fminf(mn, fminf(fminf(f.x, f.y), fminf(f.z, f.w)));
        mx = fmaxf(mx, fmaxf(fmaxf(f.x, f.y), fmaxf(f.z, f.w)));
    }
    smn[t] = mn; smx[t] = mx;
    __syncthreads();
    for (int s = 128; s > 0; s >>= 1) {
        if (t < s) {
            smn[t] = fminf(smn[t], smn[t + s]);
            smx[t] = fmaxf(smx[t], smx[t + s]);
        }
        __syncthreads();
    }
    if (t == 0) { wsmin[blockIdx.x] = smn[0]; wsmax[blockIdx.x] = smx[0]; }
}

// ---------------- kernel 2: final min/max + candidate params ----------------
__global__ __launch_bounds__(512) void prep_params(float* __restrict__ ws) {
    __shared__ float smn[512], smx[512];
    int t = threadIdx.x;
    smn[t] = ws[OFF_BMIN + t];
    smx[t] = ws[OFF_BMAX + t];
    __syncthreads();
    for (int s = 256; s > 0; s >>= 1) {
        if (t < s) {
            smn[t] = fminf(smn[t], smn[t + s]);
            smx[t] = fmaxf(smx[t], smx[t + s]);
        }
        __syncthreads();
    }
    float xmin = smn[0], xmax = smx[0];
    if (t == 0) { ws[OFF_MM] = xmin; ws[OFF_MM + 1] = xmax; }
    float xr = xmax - xmin;
    for (int c = t; c < NCAND; c += 512) {
        int   i    = (c >> 4) + 1;          // 1..100
        float z    = (float)(c & 15);       // 0..15
        float tmax = xr / 100.0f * (float)i;
        float tdel = tmax / 15.0f;
        float nmin = fmaxf(-z * tdel, xmin);
        float nmax = fminf(tmax - z * tdel, xmax);
        float mneg = fminf(nmin, 0.0f);
        float mpos = fmaxf(nmax, 0.0f);
        float scale = fmaxf((mpos - mneg) / 15.0f, EPSF);
        float zp = fminf(fmaxf(-rintf(mneg / scale), 0.0f), 15.0f);
        ws[OFF_S + c] = scale;
        ws[OFF_Z + c] = zp;
    }
}

// ---------------- kernel 3: LDS-tiled candidate scoring ----------------
__global__ __launch_bounds__(256) void score_part(const float* __restrict__ x,
                                                  const float* __restrict__ ws,
                                                  float* __restrict__ part) {
    __shared__ float tile[TILE];
    int t = threadIdx.x;
    const float* gp = x + (size_t)blockIdx.x * TILE;

#if defined(HAVE_ASYNC_LDS)
    // gfx1250 async global->LDS staging (ASYNCcnt-tracked, no VGPR round-trip)
    float* gq = const_cast<float*>(gp);
#pragma unroll
    for (int v = 0; v < 8; ++v) {
        int fi = v * 256 + t;   // float4 index
        __builtin_amdgcn_global_load_async_to_lds_b128(
            (gv4i*)(gq + fi * 4),
            (lv4i*)&tile[fi * 4],
            0, 0);
    }
#if __has_builtin(__builtin_amdgcn_s_wait_asynccnt)
    __builtin_amdgcn_s_wait_asynccnt(0);
#else
    asm volatile("s_wait_asynccnt 0" ::: "memory");
#endif
#else
    const float4* x4 = (const float4*)gp;
    float4* t4 = (float4*)tile;
#pragma unroll
    for (int v = 0; v < 8; ++v) {
        int fi = v * 256 + t;
        t4[fi] = x4[fi];
    }
#endif
    __syncthreads();

    float sc[CPT], zz[CPT], acc[CPT];
#pragma unroll
    for (int q = 0; q < CPT; ++q) {
        int c = t + q * 256;
        bool ok = c < NCAND;
        sc[q] = ok ? ws[OFF_S + c] : 1.0f;
        zz[q] = ok ? ws[OFF_Z + c] : 0.0f;
        acc[q] = 0.0f;
    }
    for (int e = 0; e < TILE; ++e) {
        float xv = tile[e];        // LDS broadcast: all lanes read same address
#pragma unroll
        for (int q = 0; q < CPT; ++q) {
            float r  = rintf(xv / sc[q]) + zz[q];
            r        = fminf(fmaxf(r, 0.0f), 15.0f);
            float d  = (r - zz[q]) * sc[q] - xv;
            acc[q]   = fmaf(d, d, acc[q]);
        }
    }
#pragma unroll
    for (int q = 0; q < CPT; ++q) {
        int c = t + q * 256;
        if (c < NCAND) part[(size_t)c * NBLK + blockIdx.x] = acc[q];
    }
}

// ---------------- kernel 4: WMMA ones-reduction of partials ----------------
// scores[c] = (1/N) * sum_b part[c][b].  One wave per group of 16 candidates.
// D = A(16x4 partials) x B(4x16 ones) + C  -> row sums, exact fp32 accumulate.
__global__ __launch_bounds__(32) void score_reduce(const float* __restrict__ part,
                                                   float* __restrict__ scores) {
    int g     = blockIdx.x;       // 0..99
    int lane  = threadIdx.x;      // 0..31
    int m     = lane & 15;        // matrix row = candidate within group
    int khalf = lane >> 4;        // which K-pair this lane supplies
    const float* row = part + (size_t)(g * 16 + m) * NBLK;
    const float inv = 1.0f / (float)NELEM;

#if defined(HAVE_WMMA_F32X4)
    v8f accv = {};
    v2f ones; ones[0] = 1.0f; ones[1] = 1.0f;
    for (int kb = 0; kb < NBLK / 4; ++kb) {
        int k0 = kb * 4 + khalf * 2;
        v2f a; a[0] = row[k0]; a[1] = row[k0 + 1];
        accv = __builtin_amdgcn_wmma_f32_16x16x4_f32(
            false, a, false, ones, (short)0, accv, false, false);
    }
    // D layout: VGPR r = row (r + 8*khalf), value replicated over all 16 columns.
    if (m == 0) {  // lanes 0 and 16 emit rows 0-7 and 8-15
        int base = g * 16 + khalf * 8;
#pragma unroll
        for (int r = 0; r < 8; ++r) scores[base + r] = accv[r] * inv;
    }
#else
    if (lane < 16) {
        float s = 0.0f;
        for (int b = 0; b < NBLK; ++b) s += row[b];
        scores[g * 16 + m] = s * inv;
    }
#endif
}

// ---------------- kernel 5: argmin with reference tie-breaking ----------------
__global__ void select_best(const float* __restrict__ ws, float* __restrict__ out) {
    if (threadIdx.x != 0) return;
    float xmin = ws[OFF_MM], xmax = ws[OFF_MM + 1];
    float best = 1.0e10f;
    int bi = 1, bj = 0;
    for (int i = 1; i <= 100; ++i) {
        float sc = INFINITY; int j = 0;
        for (int z = 0; z < 16; ++z) {
            float v = ws[OFF_SCORES + (i - 1) * 16 + z];
            if (v < sc) { sc = v; j = z; }      // first-index argmin
        }
        if (sc < best) { best = sc; bi = i; bj = j; }  // strict improvement
    }
    float xr   = xmax - xmin;
    float tmax = xr / 100.0f * (float)bi;
    float tdel = tmax / 15.0f;
    float z    = (float)bj;
    out[0] = fmaxf(-z * tdel, xmin);          // best_min
    out[1] = fminf(tmax - z * tdel, xmax);    // best_max
}

extern "C" void kernel_launch(void* const* d_in, const int* in_sizes, int n_in,
                              void* d_out, int out_size, void* d_ws, size_t ws_size,
                              hipStream_t stream) {
    const float* x = (const float*)d_in[0];
    float* ws  = (float*)d_ws;
    float* out = (float*)d_out;

    minmax_part <<<NBLK, 256, 0, stream>>>(x, ws + OFF_BMIN, ws + OFF_BMAX);
    prep_params <<<1, 512, 0, stream>>>(ws);
    score_part  <<<NBLK, 256, 0, stream>>>(x, ws, ws + OFF_PART);
    score_reduce<<<NCAND / 16, 32, 0, stream>>>(ws + OFF_PART, ws + OFF_SCORES);
    select_best <<<1, 32, 0, stream>>>(ws, out);
}